// GATGraphClassifier_11553462026422
// MI455X (gfx1250) — compile-verified
//
#include <hip/hip_runtime.h>
#include <hip/hip_bf16.h>

typedef __attribute__((ext_vector_type(2))) float v2f;
typedef __attribute__((ext_vector_type(8))) float v8f;

#define TPB 256
#define NEG_SLOPE 0.2f

// ---------- ordered-uint encoding for float atomic max ----------
__device__ __forceinline__ unsigned fenc(float f) {
  unsigned u = __float_as_uint(f);
  return (u & 0x80000000u) ? ~u : (u | 0x80000000u);
}
__device__ __forceinline__ float fdec(unsigned u) {
  return __uint_as_float((u & 0x80000000u) ? (u & 0x7fffffffu) : ~u);
}

// ---------- fills ----------
__global__ void fill_f32(float* __restrict__ p, float v, int n) {
  int t = blockIdx.x * blockDim.x + threadIdx.x;
  if (t < n) p[t] = v;
}
__global__ void fill_u32(unsigned* __restrict__ p, unsigned v, int n) {
  int t = blockIdx.x * blockDim.x + threadIdx.x;
  if (t < n) p[t] = v;
}

// ---------- FP32 WMMA GEMM: C[M,Nc] = A[M,K] * B[K,Nc] ----------
// One wave per 16x64 output strip: 4 WMMA accumulators share one A operand
// per K-step (4x A reuse, 4 independent v_wmma chains per iteration).
// Requires M % 16 == 0, K % 4 == 0, Nc % 64 == 0.
__global__ __launch_bounds__(TPB)
void wmma_gemm_f32(const float* __restrict__ A, const float* __restrict__ B,
                   float* __restrict__ C, int M, int K, int Nc, int waves_total) {
  int w    = (blockIdx.x * blockDim.x + threadIdx.x) >> 5;
  int lane = threadIdx.x & 31;
  if (w >= waves_total) return;                 // wave-uniform exit (EXEC stays all-1s)
  int nstrips = Nc >> 6;                        // 64-wide strips
  int tm = w / nstrips;
  int ts = w - tm * nstrips;

  int l15   = lane & 15;
  int arow  = tm * 16 + l15;                    // A: lanes 0-15 M=lane, lanes 16-31 M=lane-16, K+2
  int khalf = (lane >> 4) * 2;

  v8f acc0 = {}, acc1 = {}, acc2 = {}, acc3 = {};
  const float* ap = A + (size_t)arow * K + khalf;
  const float* bp = B + (size_t)(ts * 64 + l15) + (size_t)khalf * Nc;

  for (int k = 0; k < K; k += 4) {
    v2f a;
    a.x = ap[k];
    a.y = ap[k + 1];
    const float* br0 = bp + (size_t)k * Nc;     // row k+khalf
    const float* br1 = br0 + Nc;                // row k+khalf+1
    v2f b0, b1, b2, b3;
    b0.x = br0[0];  b0.y = br1[0];
    b1.x = br0[16]; b1.y = br1[16];
    b2.x = br0[32]; b2.y = br1[32];
    b3.x = br0[48]; b3.y = br1[48];
    acc0 = __builtin_amdgcn_wmma_f32_16x16x4_f32(false, a, false, b0, (short)0, acc0, false, false);
    acc1 = __builtin_amdgcn_wmma_f32_16x16x4_f32(false, a, false, b1, (short)0, acc1, false, false);
    acc2 = __builtin_amdgcn_wmma_f32_16x16x4_f32(false, a, false, b2, (short)0, acc2, false, false);
    acc3 = __builtin_amdgcn_wmma_f32_16x16x4_f32(false, a, false, b3, (short)0, acc3, false, false);
  }

  int orow = tm * 16 + (lane >> 4) * 8;         // C/D: lanes 16-31 carry M+8
  float* cp = C + (size_t)orow * Nc + ts * 64 + l15;
#pragma unroll
  for (int r = 0; r < 8; ++r) {
    float* crow = cp + (size_t)r * Nc;
    crow[0]  = acc0[r];
    crow[16] = acc1[r];
    crow[32] = acc2[r];
    crow[48] = acc3[r];
  }
}

// ---------- attention coefficients: al_s/al_d [N, heads] ----------
__global__ void alpha_coef(const float* __restrict__ H,
                           const float* __restrict__ a_src, const float* __restrict__ a_dst,
                           float* __restrict__ al_s, float* __restrict__ al_d,
                           int N, int heads) {
  int t = blockIdx.x * blockDim.x + threadIdx.x;
  if (t >= N * heads) return;
  int n = t / heads, h = t - n * heads;
  const float* hp = H + (size_t)n * heads * 64 + h * 64;
  const float* as = a_src + h * 64;
  const float* ad = a_dst + h * 64;
  float ss = 0.f, sd = 0.f;
#pragma unroll 8
  for (int i = 0; i < 64; ++i) { float v = hp[i]; ss += v * as[i]; sd += v * ad[i]; }
  al_s[t] = ss;
  al_d[t] = sd;
}

__device__ __forceinline__ void edge_sd(const int* __restrict__ ei, int E, int e,
                                        int& s, int& d) {
  if (e < E) { s = ei[e]; d = ei[E + e]; }
  else       { s = d = e - E; }                 // self loop
}

// ---------- pass 1: segment max of leaky-relu attention logits ----------
__global__ void edge_max(const float* __restrict__ al_s, const float* __restrict__ al_d,
                         unsigned* __restrict__ m_enc, const int* __restrict__ ei,
                         int E, int Etot, int heads) {
  int t = blockIdx.x * blockDim.x + threadIdx.x;
  if (t >= Etot * heads) return;
  int e = t / heads, h = t - e * heads;
  int s, d; edge_sd(ei, E, e, s, d);
  float v = al_s[s * heads + h] + al_d[d * heads + h];
  v = v > 0.f ? v : NEG_SLOPE * v;
  atomicMax(&m_enc[d * heads + h], fenc(v));
}

// ---------- pass 2: exp + accumulate numerator/denominator (wave per edge*head) ----------
__global__ __launch_bounds__(TPB)
void edge_acc(const float* __restrict__ H,
              const float* __restrict__ al_s, const float* __restrict__ al_d,
              const unsigned* __restrict__ m_enc,
              float* __restrict__ denom, float* __restrict__ num,
              const int* __restrict__ ei, int E, int Etot, int heads) {
  int w    = (blockIdx.x * blockDim.x + threadIdx.x) >> 5;
  int lane = threadIdx.x & 31;
  if (w >= Etot * heads) return;
  int e = w / heads, h = w - e * heads;
  int s, d; edge_sd(ei, E, e, s, d);
  float v = al_s[s * heads + h] + al_d[d * heads + h];
  v = v > 0.f ? v : NEG_SLOPE * v;
  float ex = __expf(v - fdec(m_enc[d * heads + h]));
  if (lane == 0) atomicAdd(&denom[d * heads + h], ex);
  int HD = heads * 64;
  const float* hs = H + (size_t)s * HD + h * 64;
  float* nd = num + (size_t)d * HD + h * 64;
  atomicAdd(&nd[lane],      ex * hs[lane]);
  atomicAdd(&nd[lane + 32], ex * hs[lane + 32]);
}

// ---------- normalize + bias + ELU (in place) ----------
__global__ void finalize_elu(float* __restrict__ num, const float* __restrict__ denom,
                             const float* __restrict__ bias, int N, int heads) {
  int HD = heads * 64;
  int t = blockIdx.x * blockDim.x + threadIdx.x;
  if (t >= N * HD) return;
  int n = t / HD, c = t - n * HD, h = c >> 6;
  float v = num[t] / (denom[n * heads + h] + 1e-16f) + bias[c];
  num[t] = v > 0.f ? v : (__expf(v) - 1.f);
}

// ---------- global mean pool (accumulate) ----------
__global__ void pool_acc(const float* __restrict__ H2, const int* __restrict__ batch,
                         float* __restrict__ pooled, float* __restrict__ counts, int N) {
  int t = blockIdx.x * blockDim.x + threadIdx.x;
  if (t >= N * 64) return;
  int n = t >> 6, dd = t & 63;
  int g = batch[n];
  atomicAdd(&pooled[g * 64 + dd], H2[t]);
  if (dd == 0) atomicAdd(&counts[g], 1.f);
}

// ---------- classifier ----------
__global__ void classify(const float* __restrict__ pooled, const float* __restrict__ counts,
                         const float* __restrict__ W, const float* __restrict__ b,
                         float* __restrict__ out, int G, int C) {
  int t = blockIdx.x * blockDim.x + threadIdx.x;
  if (t >= G * C) return;
  int g = t / C, c = t - g * C;
  float cnt = counts[g]; if (cnt < 1.f) cnt = 1.f;
  float s = 0.f;
#pragma unroll 8
  for (int d = 0; d < 64; ++d) s += pooled[g * 64 + d] * W[d * C + c];
  out[t] = s / cnt + b[c];
}

static inline int cdiv(long long a, long long b) { return (int)((a + b - 1) / b); }

extern "C" void kernel_launch(void* const* d_in, const int* in_sizes, int n_in,
                              void* d_out, int out_size, void* d_ws, size_t ws_size,
                              hipStream_t stream) {
  const float* x     = (const float*)d_in[0];
  const int*   ei    = (const int*)  d_in[1];
  const int*   batch = (const int*)  d_in[2];
  const float* W1    = (const float*)d_in[3];
  const float* a1s   = (const float*)d_in[4];
  const float* a1d   = (const float*)d_in[5];
  const float* b1    = (const float*)d_in[6];
  const float* W2    = (const float*)d_in[7];
  const float* a2s   = (const float*)d_in[8];
  const float* a2d   = (const float*)d_in[9];
  const float* b2    = (const float*)d_in[10];
  const float* linW  = (const float*)d_in[11];
  const float* linb  = (const float*)d_in[12];
  float* out = (float*)d_out;

  const int N      = in_sizes[2];            // 50000 (batch vector length)
  const int DIN    = in_sizes[0] / N;        // 128
  const int E      = in_sizes[1] / 2;        // 800000
  const int HID    = 64;
  const int HEADS1 = in_sizes[4] / HID;      // 4
  const int HD1    = HEADS1 * HID;           // 256
  const int C      = in_sizes[12];           // 10
  const int G      = out_size / C;           // 128
  const int Etot   = E + N;

  char* ws = (char*)d_ws;
  size_t off = 0;
  auto alloc = [&](size_t bytes) -> void* {
    void* p = ws + off;
    off = (off + bytes + 255) & ~(size_t)255;
    return p;
  };

  float*    H1    = (float*)   alloc((size_t)N * HD1 * 4);   // x @ W1
  float*    num1  = (float*)   alloc((size_t)N * HD1 * 4);   // numerator -> X2 (in place)
  float*    als1  = (float*)   alloc((size_t)N * HEADS1 * 4);
  float*    ald1  = (float*)   alloc((size_t)N * HEADS1 * 4);
  unsigned* m1    = (unsigned*)alloc((size_t)N * HEADS1 * 4);
  float*    den1  = (float*)   alloc((size_t)N * HEADS1 * 4);
  float*    H2    = (float*)   alloc((size_t)N * HID * 4);   // X2 @ W2
  float*    num2  = (float*)   alloc((size_t)N * HID * 4);   // numerator -> h2 (in place)
  float*    als2  = (float*)   alloc((size_t)N * 4);
  float*    ald2  = (float*)   alloc((size_t)N * 4);
  unsigned* m2    = (unsigned*)alloc((size_t)N * 4);
  float*    den2  = (float*)   alloc((size_t)N * 4);
  float*    pooled= (float*)   alloc((size_t)G * HID * 4);
  float*    cnts  = (float*)   alloc((size_t)G * 4);

  const unsigned NEGINF = 0x00000000u; // smallest ordered code (= fenc of most-negative)
  // ---- zero / init accumulators (must be re-done every call) ----
  fill_f32<<<cdiv((long long)N * HD1, TPB), TPB, 0, stream>>>(num1, 0.f, N * HD1);
  fill_f32<<<cdiv(N * HEADS1, TPB), TPB, 0, stream>>>(den1, 0.f, N * HEADS1);
  fill_u32<<<cdiv(N * HEADS1, TPB), TPB, 0, stream>>>(m1, NEGINF, N * HEADS1);
  fill_f32<<<cdiv((long long)N * HID, TPB), TPB, 0, stream>>>(num2, 0.f, N * HID);
  fill_f32<<<cdiv(N, TPB), TPB, 0, stream>>>(den2, 0.f, N);
  fill_u32<<<cdiv(N, TPB), TPB, 0, stream>>>(m2, NEGINF, N);
  fill_f32<<<cdiv(G * HID, TPB), TPB, 0, stream>>>(pooled, 0.f, G * HID);
  fill_f32<<<cdiv(G, TPB), TPB, 0, stream>>>(cnts, 0.f, G);

  // ---- layer 1 ----
  {
    int waves = (N / 16) * (HD1 / 64);         // 16x64 strips
    wmma_gemm_f32<<<cdiv((long long)waves * 32, TPB), TPB, 0, stream>>>(
        x, W1, H1, N, DIN, HD1, waves);
    alpha_coef<<<cdiv(N * HEADS1, TPB), TPB, 0, stream>>>(H1, a1s, a1d, als1, ald1, N, HEADS1);
    edge_max<<<cdiv((long long)Etot * HEADS1, TPB), TPB, 0, stream>>>(
        als1, ald1, m1, ei, E, Etot, HEADS1);
    edge_acc<<<cdiv((long long)Etot * HEADS1 * 32, TPB), TPB, 0, stream>>>(
        H1, als1, ald1, m1, den1, num1, ei, E, Etot, HEADS1);
    finalize_elu<<<cdiv((long long)N * HD1, TPB), TPB, 0, stream>>>(num1, den1, b1, N, HEADS1);
  }

  // ---- layer 2 (input = num1 == X2) ----
  {
    int waves = (N / 16) * (HID / 64);         // 16x64 strips
    wmma_gemm_f32<<<cdiv((long long)waves * 32, TPB), TPB, 0, stream>>>(
        num1, W2, H2, N, HD1, HID, waves);
    alpha_coef<<<cdiv(N, TPB), TPB, 0, stream>>>(H2, a2s, a2d, als2, ald2, N, 1);
    edge_max<<<cdiv(Etot, TPB), TPB, 0, stream>>>(als2, ald2, m2, ei, E, Etot, 1);
    edge_acc<<<cdiv((long long)Etot * 32, TPB), TPB, 0, stream>>>(
        H2, als2, ald2, m2, den2, num2, ei, E, Etot, 1);
    finalize_elu<<<cdiv((long long)N * HID, TPB), TPB, 0, stream>>>(num2, den2, b2, N, 1);
  }

  // ---- pool + classify ----
  pool_acc<<<cdiv((long long)N * HID, TPB), TPB, 0, stream>>>(num2, batch, pooled, cnts, N);
  classify<<<cdiv(G * C, TPB), TPB, 0, stream>>>(pooled, cnts, linW, linb, out, G, C);
}